// BboxProposal_48696339202426
// MI455X (gfx1250) — compile-verified
//
#include <hip/hip_runtime.h>
#include <stdint.h>

#define N_BOX    4096
#define NTHREADS 1024
#define KEEP_MAX 300
#define NMS_THR  0.5f

// Fused per-image kernel: u64-key LDS bitonic sort (stable desc-score order),
// sorted-box gather into LDS, wave32 greedy NMS with __any ballot, padded
// float4 output. One workgroup per image; ~102KB LDS (needs CDNA5 320KB WGP LDS).
__global__ __launch_bounds__(NTHREADS, 1)
void BboxProposal_nms_kernel(const float* __restrict__ cls,
                             const float* __restrict__ boxes,
                             float* __restrict__ out)
{
    __shared__ unsigned long long skey[N_BOX];   // 32 KB: sort keys
    __shared__ float4 sbox[N_BOX];               // 64 KB: boxes in sorted order
    __shared__ float4 kept[KEEP_MAX];            // kept box coords
    __shared__ float  karea[KEEP_MAX];           // kept box areas
    __shared__ int    s_count;

    const int img = blockIdx.x;
    const int tid = threadIdx.x;

    const float*  cimg = cls + (size_t)img * N_BOX * 2;
    const float4* bimg = (const float4*)(boxes + (size_t)img * N_BOX * 4);

    // Prefetch the box gather region (gfx1250 global_prefetch_b8). One byte
    // per 64B line is enough to pull whole lines.
    for (int i = tid; i < N_BOX; i += NTHREADS * 4) {
        __builtin_prefetch((const void*)&bimg[i], 0, 1);
    }

    // ---- build sort keys: ascending u64 order == (score desc, index asc) ----
    for (int i = tid; i < N_BOX; i += NTHREADS) {
        float s = cimg[2 * i + 1];
        unsigned int u   = __float_as_uint(s);
        unsigned int ord = u ^ ((u >> 31) ? 0xFFFFFFFFu : 0x80000000u); // monotone map
        unsigned int hi  = ~ord;                                        // flip -> descending score
        skey[i] = ((unsigned long long)hi << 32) | (unsigned int)i;
    }
    __syncthreads();

    // ---- bitonic sort (ascending u64), 4096 elems, 1024 threads x 4 ----
    for (int k = 2; k <= N_BOX; k <<= 1) {
        for (int j = k >> 1; j > 0; j >>= 1) {
            for (int i = tid; i < N_BOX; i += NTHREADS) {
                int p = i ^ j;
                if (p > i) {
                    unsigned long long a = skey[i];
                    unsigned long long b = skey[p];
                    bool up = ((i & k) == 0);
                    if ((a > b) == up) { skey[i] = b; skey[p] = a; }
                }
            }
            __syncthreads();
        }
    }

    // ---- gather boxes into sorted order (16B ds stores / global_load_b128) ----
    for (int i = tid; i < N_BOX; i += NTHREADS) {
        unsigned int oi = (unsigned int)(skey[i] & 0xFFFFFFFFull);
        sbox[i] = bimg[oi];
    }
    if (tid == 0) s_count = 0;
    __syncthreads();

    // ---- greedy NMS on wave 0 (wave32 ballot drives the data-dependent scan) ----
    if (tid < 32) {
        const int lane = tid;
        int count = 0;
        for (int i = 0; i < N_BOX; ++i) {
            float4 c  = sbox[i];                       // same addr all lanes -> LDS broadcast
            float  ca = (c.z - c.x) * (c.w - c.y);     // (y2-y1)*(x2-x1)
            bool sup = false;
            for (int j = lane; j < count; j += 32) {
                float4 kb = kept[j];
                float ih = fminf(c.z, kb.z) - fmaxf(c.x, kb.x); // y overlap
                float iw = fminf(c.w, kb.w) - fmaxf(c.y, kb.y); // x overlap
                ih = fmaxf(ih, 0.0f);
                iw = fmaxf(iw, 0.0f);
                float inter = ih * iw;
                float uni   = ca + karea[j] - inter;
                float iou   = (uni > 0.0f) ? (inter / uni) : 0.0f;  // match reference guard
                sup = sup || (iou > NMS_THR);
            }
            if (!__any((int)sup)) {
                if (lane == 0) { kept[count] = c; karea[count] = ca; }
                __builtin_amdgcn_wave_barrier();       // keep ds_store before next iter's loads
                count++;
                if (count >= KEEP_MAX) break;          // output only uses first 300 kept
            }
        }
        if (lane == 0) s_count = count;
    }
    __syncthreads();

    // ---- padded output: [BBOX_NUM, 4] rows, -1 fill ----
    const int cnt = s_count;
    float4* o = (float4*)(out + (size_t)img * KEEP_MAX * 4);
    for (int r = tid; r < KEEP_MAX; r += NTHREADS) {
        o[r] = (r < cnt) ? kept[r] : make_float4(-1.0f, -1.0f, -1.0f, -1.0f);
    }
}

extern "C" void kernel_launch(void* const* d_in, const int* in_sizes, int n_in,
                              void* d_out, int out_size, void* d_ws, size_t ws_size,
                              hipStream_t stream)
{
    (void)n_in; (void)out_size; (void)d_ws; (void)ws_size;
    const float* cls   = (const float*)d_in[0];   // [B, 4096, 2] f32
    const float* boxes = (const float*)d_in[1];   // [B, 4096, 4] f32
    float*       out   = (float*)d_out;           // [B, 300, 4] f32

    const int B = in_sizes[0] / (N_BOX * 2);      // = 8
    BboxProposal_nms_kernel<<<B, NTHREADS, 0, stream>>>(cls, boxes, out);
}